// EdgeConv_8650064134299
// MI455X (gfx1250) — compile-verified
//
#include <hip/hip_runtime.h>
#include <math.h>

typedef float v2f __attribute__((ext_vector_type(2)));
typedef float v8f __attribute__((ext_vector_type(8)));

#define NPTS   65536
#define SNB    16
#define CCH    64
#define WAVES  8
#define NBLK   (NPTS / WAVES)
#define ESTR   68           // padded row stride for 16x64 tiles (conflict-free)
#define EPSV   1e-5f
#define NEGS   0.1f

// finalized BN affine params in d_ws (floats)
#define WS_SC1 0
#define WS_SH1 64
#define WS_SC2 128
#define WS_SH2 192
#define WS_SCP 256
#define WS_SHP 260

// per-block stat partials staged in d_out (floats); overwritten by final kernel
#define P_Z 0u              // blk*128 + o = sum, blk*128 + 64 + o = sumsq
#define P_H (1u << 20)      // blk*8 + j = h-sum (j<3), blk*8 + 4 + j = h-sumsq

__device__ __forceinline__ float leaky(float u) { return fmaxf(u, NEGS * u); }

// Pack w[o][k] (row-major, kdim cols) into LDS as float4 per k-step:
// wq[(t*64+o)*4 + c] = w[o][4t+c].  Lane-half then reads .xy or .zw of one slot.
__device__ __forceinline__ void pack_w(const float* __restrict__ w, float* wq, int ksteps, int kdim)
{
    int total = ksteps * 64;
    for (int e = threadIdx.x; e < total; e += 256) {
        int t = e >> 6, o = e & 63;
        const float* src = w + o * kdim + 4 * t;
        wq[e * 4 + 0] = src[0];
        wq[e * 4 + 1] = src[1];
        wq[e * 4 + 2] = src[2];
        wq[e * 4 + 3] = src[3];
    }
}

// Stage center row + 16 edge rows (x[idx]-x[n]) into LDS for one wave/point.
__device__ __forceinline__ void gather_tile(const float* __restrict__ x,
                                            const int* __restrict__ idx,
                                            int n, int lane,
                                            float* xc, float* edge, int* idl)
{
    int half = lane >> 4, m = lane & 15;
    float c0 = x[n * CCH + lane];
    float c1 = x[n * CCH + 32 + lane];
    xc[lane]      = c0;
    xc[32 + lane] = c1;
    if (half == 0) idl[m] = idx[n * SNB + m];
    for (int s = 0; s < SNB; ++s) {
        int j = idl[s];
        edge[s * ESTR + lane]      = x[j * CCH + lane]      - c0;
        edge[s * ESTR + 32 + lane] = x[j * CCH + 32 + lane] - c1;
    }
}

// conv1: z1(16x64) = feat(16x128) @ w1^T via V_WMMA_F32_16X16X4_F32.
// feat[:, 0:64] = broadcast center (xc), feat[:, 64:128] = edge rows.
// Split + fully unrolled so every ds_load uses an immediate offset (no per-step
// address math / cndmask in the hot loop).
__device__ __forceinline__ void conv1_wmma(const float* xc, const float* edge,
                                           const float* wq1, int half, int m, v8f acc[4])
{
    const float* xcb = xc + 2 * half;                 // center base (broadcast)
    const float* eb  = edge + m * ESTR + 2 * half;    // edge row base
    const float* wb  = wq1 + (m * 4 + 2 * half);      // weight base
#pragma unroll
    for (int t = 0; t < 16; ++t) {
        v2f a = *(const v2f*)&xcb[4 * t];
#pragma unroll
        for (int g = 0; g < 4; ++g) {
            v2f b = *(const v2f*)&wb[(t * 64 + g * 16) * 4];
            acc[g] = __builtin_amdgcn_wmma_f32_16x16x4_f32(false, a, false, b,
                                                           (short)0, acc[g], false, false);
        }
    }
#pragma unroll
    for (int t = 0; t < 16; ++t) {
        v2f a = *(const v2f*)&eb[4 * t];
#pragma unroll
        for (int g = 0; g < 4; ++g) {
            v2f b = *(const v2f*)&wb[((t + 16) * 64 + g * 16) * 4];
            acc[g] = __builtin_amdgcn_wmma_f32_16x16x4_f32(false, a, false, b,
                                                           (short)0, acc[g], false, false);
        }
    }
}

// conv2: z2(16x64) = y1(16x64) @ w2^T
__device__ __forceinline__ void conv2_wmma(const float* y, const float* wq2,
                                           int half, int m, v8f acc[4])
{
    const float* yb = y + m * ESTR + 2 * half;
    const float* wb = wq2 + (m * 4 + 2 * half);
#pragma unroll
    for (int t = 0; t < 16; ++t) {
        v2f a = *(const v2f*)&yb[4 * t];
#pragma unroll
        for (int g = 0; g < 4; ++g) {
            v2f b = *(const v2f*)&wb[(t * 64 + g * 16) * 4];
            acc[g] = __builtin_amdgcn_wmma_f32_16x16x4_f32(false, a, false, b,
                                                           (short)0, acc[g], false, false);
        }
    }
}

// Per-wave per-channel sum / sumsq over the 16 rows of the D matrix.
__device__ __forceinline__ void chan_stats(const v8f acc[4], int half, int m,
                                           float* wsum, float* wsq)
{
#pragma unroll
    for (int g = 0; g < 4; ++g) {
        float s = 0.f, q = 0.f;
#pragma unroll
        for (int j = 0; j < 8; ++j) { float v = acc[g][j]; s += v; q += v * v; }
        s += __shfl_xor(s, 16);
        q += __shfl_xor(q, 16);
        if (half == 0) { wsum[g * 16 + m] = s; wsq[g * 16 + m] = q; }
    }
}

// ---------------- Pass 1: z1 + h statistics ----------------
__global__ __launch_bounds__(256) void k_stats1(
    const float* __restrict__ p, const float* __restrict__ x, const int* __restrict__ idx,
    const float* __restrict__ w1, const float* __restrict__ lp1w, const float* __restrict__ lp1b,
    float* __restrict__ part)
{
    __shared__ __align__(16) float wq1[32 * 64 * 4];
    __shared__ __align__(16) float xc[WAVES][64];
    __shared__ __align__(16) float edge[WAVES][SNB * ESTR];
    __shared__ int   idl[WAVES][SNB];
    __shared__ float wsum[WAVES][64];
    __shared__ float wsq[WAVES][64];
    __shared__ float whs[WAVES][8];

    pack_w(w1, wq1, 32, 128);
    __syncthreads();

    const int tid = threadIdx.x, w = tid >> 5, lane = tid & 31;
    const int half = lane >> 4, m = lane & 15;
    const int n = blockIdx.x * WAVES + w;

    gather_tile(x, idx, n, lane, xc[w], edge[w], idl[w]);

    // position-MLP pre-BN activations h (3 channels), lanes 0..15 only
    float h0 = 0.f, h1 = 0.f, h2 = 0.f;
    if (half == 0) {
        int j = idl[w][m];
        float r0 = p[j * 3 + 0] - p[n * 3 + 0];
        float r1 = p[j * 3 + 1] - p[n * 3 + 1];
        float r2 = p[j * 3 + 2] - p[n * 3 + 2];
        h0 = r0 * lp1w[0] + r1 * lp1w[1] + r2 * lp1w[2] + lp1b[0];
        h1 = r0 * lp1w[3] + r1 * lp1w[4] + r2 * lp1w[5] + lp1b[1];
        h2 = r0 * lp1w[6] + r1 * lp1w[7] + r2 * lp1w[8] + lp1b[2];
    }
    float s0 = h0, s1 = h1, s2 = h2, q0 = h0 * h0, q1 = h1 * h1, q2 = h2 * h2;
#pragma unroll
    for (int off = 16; off >= 1; off >>= 1) {
        s0 += __shfl_xor(s0, off); s1 += __shfl_xor(s1, off); s2 += __shfl_xor(s2, off);
        q0 += __shfl_xor(q0, off); q1 += __shfl_xor(q1, off); q2 += __shfl_xor(q2, off);
    }
    if (lane == 0) {
        whs[w][0] = s0; whs[w][1] = s1; whs[w][2] = s2;
        whs[w][4] = q0; whs[w][5] = q1; whs[w][6] = q2;
        whs[w][3] = 0.f; whs[w][7] = 0.f;
    }

    v8f acc[4] = {{}, {}, {}, {}};
    conv1_wmma(xc[w], edge[w], wq1, half, m, acc);
    chan_stats(acc, half, m, wsum[w], wsq[w]);
    __syncthreads();

    // fixed-order cross-wave combine -> deterministic partials in d_out scratch
    if (tid < 64) {
        float s = 0.f, q = 0.f;
        for (int ww = 0; ww < WAVES; ++ww) { s += wsum[ww][tid]; q += wsq[ww][tid]; }
        part[P_Z + blockIdx.x * 128 + tid]      = s;
        part[P_Z + blockIdx.x * 128 + 64 + tid] = q;
    } else if (tid < 72) {
        int j = tid - 64;
        float s = 0.f;
        for (int ww = 0; ww < WAVES; ++ww) s += whs[ww][j];
        part[P_H + blockIdx.x * 8 + j] = s;
    }
}

// ---------------- Finalize BN1 + pos-BN ----------------
__global__ void k_fin1(const float* __restrict__ part,
                       const float* __restrict__ g1, const float* __restrict__ b1,
                       const float* __restrict__ gp, const float* __restrict__ bp,
                       float* __restrict__ wsf)
{
    const int t = threadIdx.x;
    const float inv = 1.0f / (float)((long)NPTS * SNB);
    if (t < 64) {
        float s = 0.f, q = 0.f;
        for (int blk = 0; blk < NBLK; ++blk) {
            s += part[P_Z + (unsigned)blk * 128 + t];
            q += part[P_Z + (unsigned)blk * 128 + 64 + t];
        }
        float mean = s * inv, var = q * inv - mean * mean;
        float sc = g1[t] / sqrtf(var + EPSV);
        wsf[WS_SC1 + t] = sc;
        wsf[WS_SH1 + t] = b1[t] - mean * sc;
    } else if (t < 67) {
        int j = t - 64;
        float s = 0.f, q = 0.f;
        for (int blk = 0; blk < NBLK; ++blk) {
            s += part[P_H + (unsigned)blk * 8 + j];
            q += part[P_H + (unsigned)blk * 8 + 4 + j];
        }
        float mean = s * inv, var = q * inv - mean * mean;
        float sc = gp[j] / sqrtf(var + EPSV);
        wsf[WS_SCP + j] = sc;
        wsf[WS_SHP + j] = bp[j] - mean * sc;
    }
}

// ---------------- Pass 2: z2 statistics ----------------
__global__ __launch_bounds__(256) void k_stats2(
    const float* __restrict__ x, const int* __restrict__ idx,
    const float* __restrict__ w1, const float* __restrict__ w2,
    const float* __restrict__ wsf, float* __restrict__ part)
{
    __shared__ __align__(16) float wq1[32 * 64 * 4];
    __shared__ __align__(16) float wq2[16 * 64 * 4];
    __shared__ __align__(16) float xc[WAVES][64];
    __shared__ __align__(16) float edge[WAVES][SNB * ESTR];
    __shared__ int   idl[WAVES][SNB];
    __shared__ float wsum[WAVES][64];
    __shared__ float wsq[WAVES][64];

    pack_w(w1, wq1, 32, 128);
    pack_w(w2, wq2, 16, 64);
    __syncthreads();

    const int tid = threadIdx.x, w = tid >> 5, lane = tid & 31;
    const int half = lane >> 4, m = lane & 15;
    const int n = blockIdx.x * WAVES + w;

    gather_tile(x, idx, n, lane, xc[w], edge[w], idl[w]);

    v8f acc1[4] = {{}, {}, {}, {}};
    conv1_wmma(xc[w], edge[w], wq1, half, m, acc1);

    float sc1[4], sh1[4];
#pragma unroll
    for (int g = 0; g < 4; ++g) {
        sc1[g] = wsf[WS_SC1 + g * 16 + m];
        sh1[g] = wsf[WS_SH1 + g * 16 + m];
    }
    // y1 = leaky(bn1(z1)) back to LDS (reuse edge tile), rows = neighbors
#pragma unroll
    for (int g = 0; g < 4; ++g)
#pragma unroll
        for (int j = 0; j < 8; ++j)
            edge[w][(j + 8 * half) * ESTR + g * 16 + m] = leaky(acc1[g][j] * sc1[g] + sh1[g]);

    v8f acc2[4] = {{}, {}, {}, {}};
    conv2_wmma(edge[w], wq2, half, m, acc2);
    chan_stats(acc2, half, m, wsum[w], wsq[w]);
    __syncthreads();

    if (tid < 64) {
        float s = 0.f, q = 0.f;
        for (int ww = 0; ww < WAVES; ++ww) { s += wsum[ww][tid]; q += wsq[ww][tid]; }
        part[P_Z + blockIdx.x * 128 + tid]      = s;
        part[P_Z + blockIdx.x * 128 + 64 + tid] = q;
    }
}

// ---------------- Finalize BN2 ----------------
__global__ void k_fin2(const float* __restrict__ part,
                       const float* __restrict__ g2, const float* __restrict__ b2,
                       float* __restrict__ wsf)
{
    const int t = threadIdx.x;
    if (t >= 64) return;
    const float inv = 1.0f / (float)((long)NPTS * SNB);
    float s = 0.f, q = 0.f;
    for (int blk = 0; blk < NBLK; ++blk) {
        s += part[P_Z + (unsigned)blk * 128 + t];
        q += part[P_Z + (unsigned)blk * 128 + 64 + t];
    }
    float mean = s * inv, var = q * inv - mean * mean;
    float sc = g2[t] / sqrtf(var + EPSV);
    wsf[WS_SC2 + t] = sc;
    wsf[WS_SH2 + t] = b2[t] - mean * sc;
}

// ---------------- Pass 3: full forward + max-pool ----------------
__global__ __launch_bounds__(256) void k_final(
    const float* __restrict__ p, const float* __restrict__ x, const int* __restrict__ idx,
    const float* __restrict__ w1, const float* __restrict__ w2,
    const float* __restrict__ wsf,
    const float* __restrict__ lp1w, const float* __restrict__ lp1b,
    const float* __restrict__ lp2w, const float* __restrict__ lp2b,
    float* __restrict__ out)
{
    __shared__ __align__(16) float wq1[32 * 64 * 4];
    __shared__ __align__(16) float wq2[16 * 64 * 4];
    __shared__ __align__(16) float xc[WAVES][64];
    __shared__ __align__(16) float edge[WAVES][SNB * ESTR];
    __shared__ int   idl[WAVES][SNB];
    __shared__ float hb[WAVES][SNB * 4];   // relu(bn(h)) per neighbor, stride 4

    pack_w(w1, wq1, 32, 128);
    pack_w(w2, wq2, 16, 64);
    __syncthreads();

    const int tid = threadIdx.x, w = tid >> 5, lane = tid & 31;
    const int half = lane >> 4, m = lane & 15;
    const int n = blockIdx.x * WAVES + w;

    gather_tile(x, idx, n, lane, xc[w], edge[w], idl[w]);

    // positional branch: h -> bn -> relu into LDS (lanes 0..15, one neighbor each)
    if (half == 0) {
        int j = idl[w][m];
        float r0 = p[j * 3 + 0] - p[n * 3 + 0];
        float r1 = p[j * 3 + 1] - p[n * 3 + 1];
        float r2 = p[j * 3 + 2] - p[n * 3 + 2];
        float h0 = r0 * lp1w[0] + r1 * lp1w[1] + r2 * lp1w[2] + lp1b[0];
        float h1 = r0 * lp1w[3] + r1 * lp1w[4] + r2 * lp1w[5] + lp1b[1];
        float h2 = r0 * lp1w[6] + r1 * lp1w[7] + r2 * lp1w[8] + lp1b[2];
        hb[w][m * 4 + 0] = fmaxf(h0 * wsf[WS_SCP + 0] + wsf[WS_SHP + 0], 0.f);
        hb[w][m * 4 + 1] = fmaxf(h1 * wsf[WS_SCP + 1] + wsf[WS_SHP + 1], 0.f);
        hb[w][m * 4 + 2] = fmaxf(h2 * wsf[WS_SCP + 2] + wsf[WS_SHP + 2], 0.f);
    }

    v8f acc1[4] = {{}, {}, {}, {}};
    conv1_wmma(xc[w], edge[w], wq1, half, m, acc1);

    float sc1[4], sh1[4], sc2[4], sh2[4], lw[4][3], lb[4];
#pragma unroll
    for (int g = 0; g < 4; ++g) {
        int o = g * 16 + m;
        sc1[g] = wsf[WS_SC1 + o];  sh1[g] = wsf[WS_SH1 + o];
        sc2[g] = wsf[WS_SC2 + o];  sh2[g] = wsf[WS_SH2 + o];
        lw[g][0] = lp2w[o * 3 + 0]; lw[g][1] = lp2w[o * 3 + 1]; lw[g][2] = lp2w[o * 3 + 2];
        lb[g] = lp2b[o];
    }

#pragma unroll
    for (int g = 0; g < 4; ++g)
#pragma unroll
        for (int j = 0; j < 8; ++j)
            edge[w][(j + 8 * half) * ESTR + g * 16 + m] = leaky(acc1[g][j] * sc1[g] + sh1[g]);

    v8f acc2[4] = {{}, {}, {}, {}};
    conv2_wmma(edge[w], wq2, half, m, acc2);

    // bn2 + leaky + pos + max over neighbors
#pragma unroll
    for (int g = 0; g < 4; ++g) {
        float mv = -3.402823466e+38f;
#pragma unroll
        for (int j = 0; j < 8; ++j) {
            int M = j + 8 * half;   // neighbor index for this slot
            float u = leaky(acc2[g][j] * sc2[g] + sh2[g]);
            float pos = lb[g] + hb[w][M * 4 + 0] * lw[g][0]
                              + hb[w][M * 4 + 1] * lw[g][1]
                              + hb[w][M * 4 + 2] * lw[g][2];
            mv = fmaxf(mv, u + pos);
        }
        mv = fmaxf(mv, __shfl_xor(mv, 16));
        if (half == 0) out[n * 64 + g * 16 + m] = mv;
    }
}

extern "C" void kernel_launch(void* const* d_in, const int* in_sizes, int n_in,
                              void* d_out, int out_size, void* d_ws, size_t ws_size,
                              hipStream_t stream)
{
    (void)in_sizes; (void)n_in; (void)out_size; (void)ws_size;
    const float* p    = (const float*)d_in[0];
    const float* x    = (const float*)d_in[1];
    const int*   idx  = (const int*)d_in[2];
    const float* w1   = (const float*)d_in[3];
    const float* g1   = (const float*)d_in[4];
    const float* b1   = (const float*)d_in[5];
    const float* w2   = (const float*)d_in[6];
    const float* g2   = (const float*)d_in[7];
    const float* b2   = (const float*)d_in[8];
    const float* lp1w = (const float*)d_in[9];
    const float* lp1b = (const float*)d_in[10];
    const float* gp   = (const float*)d_in[11];
    const float* bp   = (const float*)d_in[12];
    const float* lp2w = (const float*)d_in[13];
    const float* lp2b = (const float*)d_in[14];
    float* out = (float*)d_out;
    float* wsf = (float*)d_ws;

    k_stats1<<<NBLK, 256, 0, stream>>>(p, x, idx, w1, lp1w, lp1b, out);
    k_fin1<<<1, 96, 0, stream>>>(out, g1, b1, gp, bp, wsf);
    k_stats2<<<NBLK, 256, 0, stream>>>(x, idx, w1, w2, wsf, out);
    k_fin2<<<1, 64, 0, stream>>>(out, g2, b2, wsf);
    k_final<<<NBLK, 256, 0, stream>>>(p, x, idx, w1, w2, wsf, lp1w, lp1b, lp2w, lp2b, out);
}